// TripletTagger_4853313044845
// MI455X (gfx1250) — compile-verified
//
#include <hip/hip_runtime.h>
#include <hip/hip_bf16.h>

typedef __attribute__((ext_vector_type(2))) float v2f;
typedef __attribute__((ext_vector_type(8))) float v8f;

#define HDIM 768
#define TDIM 3
#define NPAD 16
#define SDIM 512
#define BATCH 128

// ---------------------------------------------------------------------------
// Kernel 1: prediction[bs, t] = sum_h inputs[bs, h] * W[t, h] + b[t]
// One wave per 16-row tile of (B*S); full-K loop with V_WMMA_F32_16X16X4_F32.
// W is staged (zero-padded to 16 rows) in LDS; inputs stream from HBM.
// ---------------------------------------------------------------------------
__global__ __launch_bounds__(128) void tagger_gemm_wmma(
    const float* __restrict__ A, const float* __restrict__ W,
    const float* __restrict__ bias, float* __restrict__ out)
{
    __shared__ float sW[NPAD * HDIM];   // 48 KB: rows 0..2 = W, rows 3..15 = 0
    __shared__ float sB[NPAD];

    for (int i = threadIdx.x; i < NPAD * HDIM; i += blockDim.x)
        sW[i] = (i < TDIM * HDIM) ? W[i] : 0.0f;
    if (threadIdx.x < NPAD)
        sB[threadIdx.x] = (threadIdx.x < TDIM) ? bias[threadIdx.x] : 0.0f;
    __syncthreads();

    const int lane = threadIdx.x & 31;
    const int half = lane >> 4;          // 0: lanes 0-15, 1: lanes 16-31
    const int r    = lane & 15;
    const int wave = blockIdx.x * (blockDim.x >> 5) + (threadIdx.x >> 5);
    const int row0 = wave * 16;          // first (b*s) row of this tile

    // f32 16x4 A frag: lane holds K = {2*half, 2*half+1} of row (row0 + r)
    const float* aptr = A + (size_t)(row0 + r) * HDIM + 2 * half;
    // f32 4x16 B frag: lane holds K = {2*half, 2*half+1} of column r (padded W row)
    const float* bptr = sW + r * HDIM + 2 * half;

    v8f c0 = {};
    v8f c1 = {};
#pragma unroll 4
    for (int k = 0; k < HDIM; k += 8) {
        __builtin_prefetch(aptr + k + 256, 0, 1);    // global_prefetch_b8, ~1KB ahead
        v2f a0 = *(const v2f*)(aptr + k);
        v2f b0 = *(const v2f*)(bptr + k);
        v2f a1 = *(const v2f*)(aptr + k + 4);
        v2f b1 = *(const v2f*)(bptr + k + 4);
        c0 = __builtin_amdgcn_wmma_f32_16x16x4_f32(false, a0, false, b0,
                                                   (short)0, c0, false, false);
        c1 = __builtin_amdgcn_wmma_f32_16x16x4_f32(false, a1, false, b1,
                                                   (short)0, c1, false, false);
    }
    v8f c = c0 + c1;

    // C layout: VGPR v, lane -> element (M = v + 8*half, N = r). Keep N < 3.
    if (r < TDIM) {
        const float bv = sB[r];
#pragma unroll
        for (int v = 0; v < 8; ++v) {
            const int m = v + 8 * half;
            out[(size_t)(row0 + m) * TDIM + r] = c[v] + bv;
        }
    }
}

// ---------------------------------------------------------------------------
// Kernel 2: CRF negative log-likelihood over the predictions.
// One thread per batch sequence; single fused pass computes the tag-path
// score and the forward-algorithm log-partition (T = 3 states).
// ---------------------------------------------------------------------------
__device__ __forceinline__ float lse3(float x, float y, float z)
{
    const float m = fmaxf(fmaxf(x, y), z);
    return m + __logf(__expf(x - m) + __expf(y - m) + __expf(z - m));
}

__global__ __launch_bounds__(BATCH) void tagger_crf(
    const float* __restrict__ pred, const int* __restrict__ y,
    const float* __restrict__ trans, const float* __restrict__ startT,
    const float* __restrict__ stopT, float* __restrict__ loss_out)
{
    __shared__ float sT[9], sS[3], sP[3];
    __shared__ float red[BATCH];

    if (threadIdx.x < 9) sT[threadIdx.x] = trans[threadIdx.x];
    if (threadIdx.x < 3) {
        sS[threadIdx.x] = startT[threadIdx.x];
        sP[threadIdx.x] = stopT[threadIdx.x];
    }
    __syncthreads();

    const int b = threadIdx.x;
    const float* p  = pred + (size_t)b * SDIM * TDIM;
    const int*   yb = y + b * SDIM;

    const float t00 = sT[0], t01 = sT[1], t02 = sT[2];
    const float t10 = sT[3], t11 = sT[4], t12 = sT[5];
    const float t20 = sT[6], t21 = sT[7], t22 = sT[8];

    // s = 0
    float e0 = p[0], e1 = p[1], e2 = p[2];
    int   prev  = yb[0];
    float score = sS[prev] + (prev == 0 ? e0 : (prev == 1 ? e1 : e2));
    float a0 = e0 + sS[0];
    float a1 = e1 + sS[1];
    float a2 = e2 + sS[2];

    for (int s = 1; s < SDIM; ++s) {
        e0 = p[s * TDIM + 0];
        e1 = p[s * TDIM + 1];
        e2 = p[s * TDIM + 2];
        const int cur = yb[s];
        score += (cur == 0 ? e0 : (cur == 1 ? e1 : e2)) + sT[prev * TDIM + cur];
        const float n0 = lse3(a0 + t00, a1 + t10, a2 + t20) + e0;
        const float n1 = lse3(a0 + t01, a1 + t11, a2 + t21) + e1;
        const float n2 = lse3(a0 + t02, a1 + t12, a2 + t22) + e2;
        a0 = n0; a1 = n1; a2 = n2;
        prev = cur;
    }
    score += sP[prev];
    const float logz = lse3(a0 + sP[0], a1 + sP[1], a2 + sP[2]);

    red[b] = score - logz;
    __syncthreads();
    if (b == 0) {
        float acc = 0.0f;
        for (int i = 0; i < BATCH; ++i) acc += red[i];
        loss_out[0] = -acc / (float)BATCH;
    }
}

// ---------------------------------------------------------------------------
extern "C" void kernel_launch(void* const* d_in, const int* in_sizes, int n_in,
                              void* d_out, int out_size, void* d_ws, size_t ws_size,
                              hipStream_t stream)
{
    (void)in_sizes; (void)n_in; (void)d_ws; (void)ws_size;

    const float* inputs = (const float*)d_in[0];   // (128, 512, 768) f32
    const int*   y      = (const int*)d_in[1];     // (128, 512) int
    const float* W      = (const float*)d_in[2];   // (3, 768) f32
    const float* b      = (const float*)d_in[3];   // (3,) f32
    const float* trans  = (const float*)d_in[4];   // (3, 3) f32
    const float* startT = (const float*)d_in[5];   // (3,) f32
    const float* stopT  = (const float*)d_in[6];   // (3,) f32

    float* out  = (float*)d_out;                   // 196608 preds + 1 loss
    float* loss = out + (out_size - 1);

    const int rows   = BATCH * SDIM;               // 65536
    const int waves  = rows / 16;                  // 4096 row tiles
    const int blocks = waves / 4;                  // 4 waves (128 thr) per block

    tagger_gemm_wmma<<<blocks, 128, 0, stream>>>(inputs, W, b, out);
    tagger_crf<<<1, BATCH, 0, stream>>>(out, y, trans, startT, stopT, loss);
}